// Model_11888469475964
// MI455X (gfx1250) — compile-verified
//
#include <hip/hip_runtime.h>
#include <cstdint>

typedef float v2f __attribute__((ext_vector_type(2)));

// Problem constants (fixed by the reference).
#define Cc 8
#define Dd 16
#define Hh 256
#define Ww 256
// 8 (c,dp)-tiles in flight per wave; slot index == c. 32 async b64 loads
// (8 KB) outstanding per wave; ASYNCcnt max is 63 so this is safe.
#define NBUF 8

__global__ __launch_bounds__(128, 1) void fused_pool_unpool_softmax(
    const float* __restrict__ x, const float* __restrict__ scale,
    const float* __restrict__ bias, float* __restrict__ out) {
  // 4 waves * 8 slots * 4 rows * 32 lanes * 8B = 32 KB
  __shared__ v2f sbuf[4 * NBUF * 4 * 32];

  const int tid  = threadIdx.x;        // wp in 0..127
  const int wave = tid >> 5;
  const int lane = tid & 31;
  const int n    = blockIdx.x >> 7;    // 4 * 128 blocks
  const int hp   = blockIdx.x & 127;

  // Global byte address of x[n, 0, 0, 2*hp, 2*wp]
  const uint64_t xbase =
      (uint64_t)(uintptr_t)x +
      (uint64_t)((n * Cc * Dd * Hh + 2 * hp) * Ww + 2 * tid) * 4ull;

  // LDS byte address of this lane's region (generic LDS addr low 32 bits ==
  // LDS offset on gfx1250: {aperture_hi32, lds_offset32}).
  const unsigned lds_base =
      (unsigned)(uintptr_t)(sbuf) + (unsigned)(wave * 8192 + lane * 8);

  // Issue one (c, dp) tile: 4 rows (ld,lh), each lane async-loads its own 8B
  // window segment directly into LDS (ASYNCcnt-tracked, wave-private).
  auto issue_tile = [&](int c, int dpx) {
#pragma unroll
    for (int r = 0; r < 4; ++r) {
      const int ld = r >> 1, lh = r & 1;
      const uint64_t ga =
          xbase + (uint64_t)(((c * Dd + 2 * dpx + ld) * Hh + lh) * Ww) * 4ull;
      const unsigned la = lds_base + (unsigned)(c * 1024 + r * 256);
      asm volatile("global_load_async_to_lds_b64 %0, %1, off"
                   :: "v"(la), "v"(ga) : "memory");
    }
  };

  // Prologue: fill the pipeline with dp=0 tiles for all 8 channels.
#pragma unroll
  for (int c = 0; c < Cc; ++c) issue_tile(c, 0);

  float sc[Cc], bs[Cc];
#pragma unroll
  for (int c = 0; c < Cc; ++c) { sc[c] = scale[c]; bs[c] = bias[c]; }

  float a[Cc][4];
#pragma unroll
  for (int c = 0; c < Cc; ++c)
#pragma unroll
    for (int q = 0; q < 4; ++q) a[c][q] = 0.0f;

  for (int dp = 0; dp < 8; ++dp) {
#pragma unroll
    for (int c = 0; c < Cc; ++c) {
      // Async loads complete in order: <=28 outstanding ==> oldest tile done.
      asm volatile("s_wait_asynccnt 28" ::: "memory");
      const int base = (wave * NBUF + c) * 4 * 32 + lane;
      v2f r0 = sbuf[base + 0 * 32];
      v2f r1 = sbuf[base + 1 * 32];
      v2f r2 = sbuf[base + 2 * 32];
      v2f r3 = sbuf[base + 3 * 32];
      // DS reads must land before async writes may reuse this slot.
      asm volatile("s_wait_dscnt 0" ::: "memory");
      if (dp < 7) issue_tile(c, dp + 1);

      // Threshold (x > 0 ? x : -0.1), window order i = ld*4 + lh*2 + lw.
      float v0 = r0.x > 0.f ? r0.x : -0.1f;  // ld0 lh0 lw0 -> q0
      float v1 = r0.y > 0.f ? r0.y : -0.1f;  // q1
      float v2 = r1.x > 0.f ? r1.x : -0.1f;  // ld0 lh1 -> q2
      float v3 = r1.y > 0.f ? r1.y : -0.1f;  // q3
      float v4 = r2.x > 0.f ? r2.x : -0.1f;  // ld1 lh0 -> q0
      float v5 = r2.y > 0.f ? r2.y : -0.1f;  // q1
      float v6 = r3.x > 0.f ? r3.x : -0.1f;  // ld1 lh1 -> q2
      float v7 = r3.y > 0.f ? r3.y : -0.1f;  // q3

      // First-occurrence argmax (strict >) matches jnp.argmax tie-breaking.
      float m = v0; int q = 0;
      if (v1 > m) { m = v1; q = 1; }
      if (v2 > m) { m = v2; q = 2; }
      if (v3 > m) { m = v3; q = 3; }
      if (v4 > m) { m = v4; q = 0; }
      if (v5 > m) { m = v5; q = 1; }
      if (v6 > m) { m = v6; q = 2; }
      if (v7 > m) { m = v7; q = 3; }

      const float scaled = fmaf(m, sc[c], bs[c]);
      a[c][0] += (q == 0) ? scaled : 0.0f;
      a[c][1] += (q == 1) ? scaled : 0.0f;
      a[c][2] += (q == 2) ? scaled : 0.0f;
      a[c][3] += (q == 3) ? scaled : 0.0f;
    }
  }

  // Mean over D (x 1/16), then softmax over channels at each of the 4
  // spatial positions (zeros elsewhere contribute exp(0)=1, as in reference).
#pragma unroll
  for (int c = 0; c < Cc; ++c)
#pragma unroll
    for (int q = 0; q < 4; ++q) a[c][q] *= 0.0625f;

#pragma unroll
  for (int q = 0; q < 4; ++q) {
    float mx = a[0][q];
#pragma unroll
    for (int c = 1; c < Cc; ++c) mx = fmaxf(mx, a[c][q]);
    float s = 0.0f;
#pragma unroll
    for (int c = 0; c < Cc; ++c) {
      a[c][q] = __expf(a[c][q] - mx);
      s += a[c][q];
    }
    const float inv = 1.0f / s;
#pragma unroll
    for (int c = 0; c < Cc; ++c) a[c][q] *= inv;
  }

  // Coalesced float2 stores: out[n, c, 2hp+lh, 2wp + (0,1)]
  float* obase = out + (uint64_t)((n * Cc * Hh + 2 * hp) * Ww + 2 * tid);
#pragma unroll
  for (int c = 0; c < Cc; ++c) {
#pragma unroll
    for (int lh = 0; lh < 2; ++lh) {
      v2f o;
      o.x = a[c][2 * lh + 0];
      o.y = a[c][2 * lh + 1];
      *reinterpret_cast<v2f*>(obase + (uint64_t)((c * Hh + lh) * Ww)) = o;
    }
  }
}

extern "C" void kernel_launch(void* const* d_in, const int* in_sizes, int n_in,
                              void* d_out, int out_size, void* d_ws,
                              size_t ws_size, hipStream_t stream) {
  const float* x     = (const float*)d_in[0];
  const float* scale = (const float*)d_in[1];
  const float* bias  = (const float*)d_in[2];
  float* out         = (float*)d_out;
  (void)in_sizes; (void)n_in; (void)out_size; (void)d_ws; (void)ws_size;
  // grid = N(4) * Hp(128) blocks, 128 threads (4 waves) each; thread = wp.
  fused_pool_unpool_softmax<<<dim3(4 * 128), dim3(128), 0, stream>>>(
      x, scale, bias, out);
}